// GNN_43009802502327
// MI455X (gfx1250) — compile-verified
//
#include <hip/hip_runtime.h>

#define N_NODES  50000
#define N_EDGES  800000
#define FEAT_DIM 16
#define HIDDEN   128
#define N_LAYERS 4
#define OUT_DIM  12
#define N_GRAPHS 1024

typedef __attribute__((ext_vector_type(2))) float v2f;
typedef __attribute__((ext_vector_type(8))) float v8f;

// ---------------------------------------------------------------------------
// Dense GEMM on the CDNA5 matrix pipe: D[M,N] = act(A[M,K] @ W[K,N] + bias)
// One wave per 16x(16*CT) output strip; K stepped by 4 via
// V_WMMA_F32_16X16X4_F32 (fp32 A/B, fp32 acc -> exact fp32 result).
// REQUIREMENTS (enforced by caller): M%16==0, K%4==0, N%(16*CT)==0.
// Branch-free inner loop: EXEC stays all-ones (WMMA requirement) and the
// compiler emits no saveexec diamonds around the loads.
//
// ISA VGPR layouts (CDNA5 ISA):
//   A 16x4 : lane r=lane&15 is row M=r; koff=(lane>>4)*2; {v0,v1}={K=koff,koff+1}
//   B 4x16 : lane n=lane&15 is col N=n; same K split across lane halves
//   C/D    : VGPR v: lanes0-15 -> M=v, N=lane; lanes16-31 -> M=v+8, N=lane-16
// ---------------------------------------------------------------------------
template <int CT>
__global__ __launch_bounds__(32)
void gemm_wmma_f32(const float* __restrict__ A, const float* __restrict__ W,
                   const float* __restrict__ bias, float* __restrict__ D,
                   int K, int N, int relu)
{
    const int lane    = threadIdx.x;              // 0..31, one wave per block
    const int rowBase = blockIdx.x * 16;
    const int colBase = blockIdx.y * (16 * CT);
    const int r    = lane & 15;
    const int koff = (lane >> 4) * 2;             // 0 or 2
    const int arow = rowBase + r;

    // A row pointer: the two K-values per lane are contiguous -> float2 (b64)
    const v2f* __restrict__ Arow = (const v2f*)(A + (size_t)arow * K);

    v8f acc[CT];
#pragma unroll
    for (int t = 0; t < CT; ++t) acc[t] = (v8f){};

    for (int kk = 0; kk < K; kk += 4) {
        const int k0 = kk + koff;
        const v2f a = Arow[k0 >> 1];              // A[arow][k0], A[arow][k0+1]
        const float* __restrict__ Wr0 = W + (size_t)k0 * N;
        const float* __restrict__ Wr1 = Wr0 + N;
#pragma unroll
        for (int t = 0; t < CT; ++t) {
            v2f b;
            const int col = colBase + t * 16 + r;
            b.x = Wr0[col];
            b.y = Wr1[col];
            acc[t] = __builtin_amdgcn_wmma_f32_16x16x4_f32(
                         false, a, false, b, (short)0, acc[t], false, false);
        }
    }

    const int n    = lane & 15;
    const int moff = (lane >> 4) * 8;
#pragma unroll
    for (int t = 0; t < CT; ++t) {
        const int ocol = colBase + t * 16 + n;
        const float bv = bias ? bias[ocol] : 0.0f;
#pragma unroll
        for (int v = 0; v < 8; ++v) {
            const int orow = rowBase + v + moff;
            float val = acc[t][v] + bv;
            if (relu) val = fmaxf(val, 0.0f);
            D[(size_t)orow * N + ocol] = val;
        }
    }
}

// ---------------------------------------------------------------------------
// Elementwise / scatter helpers (HBM + atomic bound part)
// ---------------------------------------------------------------------------
__global__ void k_fill(float* __restrict__ p, float v, int n) {
    int i = blockIdx.x * blockDim.x + threadIdx.x;
    if (i < n) p[i] = v;
}

__global__ void k_count_deg(const int* __restrict__ dst, float* __restrict__ deg, int E) {
    int e = blockIdx.x * blockDim.x + threadIdx.x;
    if (e < E) atomicAdd(&deg[dst[e]], 1.0f);
}

__global__ void k_rsqrt(float* __restrict__ d, int n) {
    int i = blockIdx.x * blockDim.x + threadIdx.x;
    if (i < n) d[i] = rsqrtf(d[i]);
}

// agg[i,:] = hw[i,:] * dinv[i]^2   (self-loop term; initializes agg)
__global__ void k_self_init(const float* __restrict__ hw, const float* __restrict__ dinv,
                            float* __restrict__ agg, int nNodes) {
    int t = blockIdx.x * blockDim.x + threadIdx.x;
    int node = t >> 5, lane = t & 31;             // 32 lanes x float4 = 128 feats
    if (node >= nNodes) return;
    float s = dinv[node]; s = s * s;
    float4 v = ((const float4*)hw)[(size_t)node * 32 + lane];
    v.x *= s; v.y *= s; v.z *= s; v.w *= s;
    ((float4*)agg)[(size_t)node * 32 + lane] = v;
}

// agg[dst,:] += hw[src,:] * dinv[src]*dinv[dst]   (one wave per edge row)
__global__ void k_edge_scatter(const int* __restrict__ src, const int* __restrict__ dst,
                               const float* __restrict__ dinv, const float* __restrict__ hw,
                               float* __restrict__ agg, int E) {
    long long t = (long long)blockIdx.x * blockDim.x + threadIdx.x;
    int e = (int)(t >> 5);
    int lane = (int)(t & 31);
    if (e >= E) return;
    int s = src[e], d = dst[e];
    float norm = dinv[s] * dinv[d];
    float4 v = ((const float4*)hw)[(size_t)s * 32 + lane];
    float* o = agg + (size_t)d * 128 + lane * 4;
    atomicAdd(o + 0, v.x * norm);
    atomicAdd(o + 1, v.y * norm);
    atomicAdd(o + 2, v.z * norm);
    atomicAdd(o + 3, v.w * norm);
}

__global__ void k_bias_relu(float* __restrict__ h, const float* __restrict__ bias, int nNodes) {
    int t = blockIdx.x * blockDim.x + threadIdx.x;
    int node = t >> 5, lane = t & 31;
    if (node >= nNodes) return;
    float4 b = ((const float4*)bias)[lane];
    float4 v = ((float4*)h)[(size_t)node * 32 + lane];
    v.x = fmaxf(v.x + b.x, 0.0f);
    v.y = fmaxf(v.y + b.y, 0.0f);
    v.z = fmaxf(v.z + b.z, 0.0f);
    v.w = fmaxf(v.w + b.w, 0.0f);
    ((float4*)h)[(size_t)node * 32 + lane] = v;
}

__global__ void k_pool_scatter(const int* __restrict__ batch, const float* __restrict__ h,
                               float* __restrict__ sums, float* __restrict__ cnt, int nNodes) {
    int t = blockIdx.x * blockDim.x + threadIdx.x;
    int node = t >> 5, lane = t & 31;
    if (node >= nNodes) return;
    int b = batch[node];
    float4 v = ((const float4*)h)[(size_t)node * 32 + lane];
    float* o = sums + (size_t)b * 128 + lane * 4;
    atomicAdd(o + 0, v.x);
    atomicAdd(o + 1, v.y);
    atomicAdd(o + 2, v.z);
    atomicAdd(o + 3, v.w);
    if (lane == 0) atomicAdd(&cnt[b], 1.0f);
}

__global__ void k_pool_div(const float* __restrict__ sums, const float* __restrict__ cnt,
                           float* __restrict__ pooled, int nGraphs) {
    int t = blockIdx.x * blockDim.x + threadIdx.x;
    int g = t >> 5, lane = t & 31;
    if (g >= nGraphs) return;
    float c = fmaxf(cnt[g], 1.0f);
    float4 v = ((const float4*)sums)[(size_t)g * 32 + lane];
    v.x /= c; v.y /= c; v.z /= c; v.w /= c;
    ((float4*)pooled)[(size_t)g * 32 + lane] = v;
}

// Pad Wo[128x12] -> WoPad[128x16] (zeros in cols 12..15) and bo -> boPad[16].
__global__ void k_pad_head(const float* __restrict__ Wo, const float* __restrict__ bo,
                           float* __restrict__ WoPad, float* __restrict__ boPad) {
    int i = blockIdx.x * blockDim.x + threadIdx.x;   // over (HIDDEN+1)*16
    if (i >= (HIDDEN + 1) * 16) return;
    int row = i >> 4, col = i & 15;
    float v = 0.0f;
    if (col < OUT_DIM) v = (row < HIDDEN) ? Wo[row * OUT_DIM + col] : bo[col];
    if (row < HIDDEN) WoPad[row * 16 + col] = v;
    else              boPad[col] = v;
}

// Compact outPad[1024x16] -> out[1024x12]
__global__ void k_compact_out(const float* __restrict__ outPad, float* __restrict__ out) {
    int i = blockIdx.x * blockDim.x + threadIdx.x;   // over N_GRAPHS*OUT_DIM
    if (i >= N_GRAPHS * OUT_DIM) return;
    int g = i / OUT_DIM, c = i % OUT_DIM;
    out[i] = outPad[g * 16 + c];
}

// ---------------------------------------------------------------------------
extern "C" void kernel_launch(void* const* d_in, const int* in_sizes, int n_in,
                              void* d_out, int out_size, void* d_ws, size_t ws_size,
                              hipStream_t stream) {
    (void)in_sizes; (void)n_in; (void)out_size; (void)ws_size;

    const float* x     = (const float*)d_in[0];
    const int*   ei    = (const int*)  d_in[1];
    const int*   src   = ei;
    const int*   dst   = ei + N_EDGES;
    const int*   batch = (const int*)  d_in[3];
    const float* We1   = (const float*)d_in[4];
    const float* be1   = (const float*)d_in[5];
    const float* We2   = (const float*)d_in[6];
    const float* be2   = (const float*)d_in[7];
    const float* Wg    = (const float*)d_in[8];
    const float* bg    = (const float*)d_in[9];
    const float* Wo    = (const float*)d_in[10];
    const float* bo    = (const float*)d_in[11];
    float*       out   = (float*)d_out;

    // Workspace layout (floats): ~52.6 MB total (fits the 192 MB L2)
    float* ws     = (float*)d_ws;
    float* h      = ws;                                    // [50000 x 128]
    float* hw     = h      + (size_t)N_NODES * HIDDEN;     // [50000 x 128]
    float* dinv   = hw     + (size_t)N_NODES * HIDDEN;     // [50000]
    float* sums   = dinv   + N_NODES;                      // [1024 x 128]
    float* cnt    = sums   + (size_t)N_GRAPHS * HIDDEN;    // [1024]
    float* pooled = cnt    + N_GRAPHS;                     // [1024 x 128]
    float* WoPad  = pooled + (size_t)N_GRAPHS * HIDDEN;    // [128 x 16]
    float* boPad  = WoPad  + HIDDEN * 16;                  // [16]
    float* outPad = boPad  + 16;                           // [1024 x 16]

    const int T = 256;
    const int nodeVec = N_NODES * 32;                      // (node, lane) threads
    const long long edgeVec = (long long)N_EDGES * 32;

    // 1) degrees: deg = 1 (self loop) + indegree;  dinv = rsqrt(deg) in place
    k_fill     <<<(N_NODES + T - 1) / T, T, 0, stream>>>(dinv, 1.0f, N_NODES);
    k_count_deg<<<(N_EDGES + T - 1) / T, T, 0, stream>>>(dst, dinv, N_EDGES);
    k_rsqrt    <<<(N_NODES + T - 1) / T, T, 0, stream>>>(dinv, N_NODES);

    // 2) node-embed MLP:  hw = relu(x@We1+be1);  h = hw@We2+be2
    dim3 gN(N_NODES / 16, HIDDEN / 64);                    // 16 x 64 strip per wave
    gemm_wmma_f32<4><<<gN, 32, 0, stream>>>(x,  We1, be1, hw, FEAT_DIM, HIDDEN, 1);
    gemm_wmma_f32<4><<<gN, 32, 0, stream>>>(hw, We2, be2, h,  HIDDEN,   HIDDEN, 0);

    // 3) GCN layers
    for (int l = 0; l < N_LAYERS; ++l) {
        gemm_wmma_f32<4><<<gN, 32, 0, stream>>>(h, Wg + (size_t)l * HIDDEN * HIDDEN,
                                                nullptr, hw, HIDDEN, HIDDEN, 0);
        k_self_init   <<<(nodeVec + T - 1) / T, T, 0, stream>>>(hw, dinv, h, N_NODES);
        k_edge_scatter<<<(int)((edgeVec + T - 1) / T), T, 0, stream>>>(src, dst, dinv, hw, h, N_EDGES);
        k_bias_relu   <<<(nodeVec + T - 1) / T, T, 0, stream>>>(h, bg + (size_t)l * HIDDEN, N_NODES);
    }

    // 4) scatter-mean pool (sums & cnt contiguous -> single zero-fill)
    k_fill<<<(N_GRAPHS * HIDDEN + N_GRAPHS + T - 1) / T, T, 0, stream>>>(
        sums, 0.0f, N_GRAPHS * HIDDEN + N_GRAPHS);
    k_pool_scatter<<<(nodeVec + T - 1) / T, T, 0, stream>>>(batch, h, sums, cnt, N_NODES);
    k_pool_div<<<(N_GRAPHS * 32 + T - 1) / T, T, 0, stream>>>(sums, cnt, pooled, N_GRAPHS);

    // 5) output head: pad N=12 -> 16 so the GEMM stays branch-free, then compact
    k_pad_head<<<((HIDDEN + 1) * 16 + T - 1) / T, T, 0, stream>>>(Wo, bo, WoPad, boPad);
    dim3 gO(N_GRAPHS / 16, 1);
    gemm_wmma_f32<1><<<gO, 32, 0, stream>>>(pooled, WoPad, boPad, outPad, HIDDEN, 16, 0);
    k_compact_out<<<(N_GRAPHS * OUT_DIM + T - 1) / T, T, 0, stream>>>(outPad, out);
}